// GraphAutoEncoder_9912784519777
// MI455X (gfx1250) — compile-verified
//
#include <hip/hip_runtime.h>

typedef __attribute__((ext_vector_type(2))) float v2f;
typedef __attribute__((ext_vector_type(8))) float v8f;

#define NNODE 8192
#define NEDGE 262144
#define VOCABS 4096
#define HID 128
#define D1 64

// Gram GEMM tiling (S = sigmoid(O @ O^T), O: [8192,4096] fp32)
#define GN 8192
#define GK 4096
#define BM 128
#define BN 128
#define BK 32
#define LDP 36   // padded LDS row stride in floats (bank-conflict free, 16B aligned)

static inline int cdiv_h(long long a, int b) { return (int)((a + b - 1) / b); }

// ---------------- small utility kernels ----------------

__global__ __launch_bounds__(256) void fill_f32(float* p, float v, long long n) {
  long long i = blockIdx.x * 256LL + threadIdx.x;
  if (i < n) p[i] = v;
}

__global__ __launch_bounds__(256) void count_deg(const int* __restrict__ dst, float* deg, int e) {
  int i = blockIdx.x * 256 + threadIdx.x;
  if (i < e) atomicAdd(&deg[dst[i]], 1.0f);
}

__global__ __launch_bounds__(256) void rsqrt_ip(float* p, int n) {
  int i = blockIdx.x * 256 + threadIdx.x;
  if (i < n) p[i] = rsqrtf(p[i]);
}

__global__ __launch_bounds__(256) void embed_k(const int* __restrict__ x,
                                               const float* __restrict__ emb,
                                               float* __restrict__ h, int n) {
  int i = blockIdx.x * 256 + threadIdx.x;
  if (i < n * HID) {
    int node = i / HID, f = i % HID;
    h[i] = emb[(long long)x[node] * HID + f];
  }
}

// C[M,Nc] = A[M,K] @ B[K,Nc] (+bias) (optional relu). One thread per output.
__global__ __launch_bounds__(256) void gemm_bias(const float* __restrict__ A,
                                                 const float* __restrict__ B,
                                                 const float* __restrict__ bias,
                                                 float* __restrict__ C,
                                                 int M, int K, int Nc, int relu) {
  long long idx = blockIdx.x * 256LL + threadIdx.x;
  if (idx >= (long long)M * Nc) return;
  int i = (int)(idx / Nc), j = (int)(idx % Nc);
  float s = bias ? bias[j] : 0.0f;
  const float* a = A + (long long)i * K;
  for (int k = 0; k < K; k++) s = fmaf(a[k], B[(long long)k * Nc + j], s);
  if (relu) s = fmaxf(s, 0.0f);
  C[idx] = s;
}

// g[dst] += norm(src,dst) * t[src]   over edges, one thread per (edge, feature)
__global__ __launch_bounds__(256) void scatter_edges(const int* __restrict__ src,
                                                     const int* __restrict__ dst,
                                                     const float* __restrict__ dinv,
                                                     const float* __restrict__ t,
                                                     float* g, int e, int d) {
  long long idx = blockIdx.x * 256LL + threadIdx.x;
  if (idx >= (long long)e * d) return;
  int ed = (int)(idx / d), f = (int)(idx % d);
  int s = src[ed], q = dst[ed];
  float nrm = dinv[s] * dinv[q];
  atomicAdd(&g[(long long)q * d + f], t[(long long)s * d + f] * nrm);
}

// g[i] = act(g[i] + t[i]*dinv[i]^2 + bias)   (fused self-loop + bias + relu)
__global__ __launch_bounds__(256) void finalize_conv(float* g, const float* __restrict__ t,
                                                     const float* __restrict__ dinv,
                                                     const float* __restrict__ bias,
                                                     int n, int d, int relu) {
  long long idx = blockIdx.x * 256LL + threadIdx.x;
  if (idx >= (long long)n * d) return;
  int i = (int)(idx / d), f = (int)(idx % d);
  float di = dinv[i];
  float v = g[idx] + t[idx] * di * di + bias[f];
  g[idx] = relu ? fmaxf(v, 0.0f) : v;
}

// ---------------- WMMA Gram + sigmoid kernel ----------------
// S[i,j] = sigmoid( sum_k O[i,k]*O[j,k] ),  fp32 WMMA 16x16x4.

__global__ __launch_bounds__(256) void gram_sigmoid_wmma(const float* __restrict__ O,
                                                         float* __restrict__ S) {
  __shared__ float At[BM * LDP];
  __shared__ float Bt[BN * LDP];

  const int i0 = blockIdx.y * BM;
  const int j0 = blockIdx.x * BN;
  const int tid = threadIdx.x;
  const int lane = tid & 31;
  const int wave = tid >> 5;      // 0..7
  const int wr = wave >> 1;       // 0..3 -> 32-row strip
  const int wc = wave & 1;        // 0..1 -> 64-col strip
  const int m_l = lane & 15;      // M/N index within a 16x16 tile
  const int k_off = (lane >> 4) << 1;  // lanes 0-15: K=0,1 ; lanes 16-31: K=2,3

  v8f acc[2][4];
  const v8f vz = {0.f, 0.f, 0.f, 0.f, 0.f, 0.f, 0.f, 0.f};
  for (int a = 0; a < 2; a++)
    for (int b = 0; b < 4; b++) acc[a][b] = vz;

  for (int k0 = 0; k0 < GK; k0 += BK) {
    // Stage A-panel (rows i0..) and B-panel (rows j0..) into LDS: 128x32 each.
    for (int s = 0; s < 4; s++) {
      int q = tid + 256 * s;          // 0..1023
      int r = q >> 3;                 // 0..127
      int c = (q & 7) << 2;           // 0,4,...,28
      *(float4*)&At[r * LDP + c] = *(const float4*)&O[(size_t)(i0 + r) * GK + k0 + c];
      *(float4*)&Bt[r * LDP + c] = *(const float4*)&O[(size_t)(j0 + r) * GK + k0 + c];
    }
    __syncthreads();

    for (int kk = 0; kk < BK / 4; kk++) {
      v2f af[2], bf[4];
      for (int mt = 0; mt < 2; mt++) {
        int row = wr * 32 + mt * 16 + m_l;
        af[mt] = *(const v2f*)&At[row * LDP + kk * 4 + k_off];
      }
      for (int nt = 0; nt < 4; nt++) {
        int col = wc * 64 + nt * 16 + m_l;
        bf[nt] = *(const v2f*)&Bt[col * LDP + kk * 4 + k_off];
      }
      for (int mt = 0; mt < 2; mt++)
        for (int nt = 0; nt < 4; nt++)
          acc[mt][nt] = __builtin_amdgcn_wmma_f32_16x16x4_f32(
              false, af[mt], false, bf[nt], (short)0, acc[mt][nt], false, false);
    }
    __syncthreads();
  }

  // Epilogue: D layout -> lane<16: (M=v, N=lane); lane>=16: (M=v+8, N=lane-16)
  const int row_half = (lane >> 4) * 8;
  for (int mt = 0; mt < 2; mt++) {
    for (int nt = 0; nt < 4; nt++) {
      int col = j0 + wc * 64 + nt * 16 + m_l;
      int rbase = i0 + wr * 32 + mt * 16 + row_half;
      for (int v = 0; v < 8; v++) {
        float val = acc[mt][nt][v];
        S[(size_t)(rbase + v) * GN + col] = 1.0f / (1.0f + __expf(-val));
      }
    }
  }
}

// ---------------- launcher ----------------

extern "C" void kernel_launch(void* const* d_in, const int* in_sizes, int n_in,
                              void* d_out, int out_size, void* d_ws, size_t ws_size,
                              hipStream_t stream) {
  const int*   x   = (const int*)d_in[0];
  const int*   ei  = (const int*)d_in[1];   // [2, E]: row0 = src, row1 = dst
  const float* emb = (const float*)d_in[2];
  const float* W1  = (const float*)d_in[3];
  const float* b1  = (const float*)d_in[4];
  const float* W2  = (const float*)d_in[5];
  const float* b2  = (const float*)d_in[6];
  const float* W3  = (const float*)d_in[7];
  const float* b3  = (const float*)d_in[8];
  const float* Wd  = (const float*)d_in[9];
  const float* bd  = (const float*)d_in[10];
  float* S = (float*)d_out;

  const int N = NNODE, E = NEDGE;
  const int* srcv = ei;
  const int* dstv = ei + E;

  // workspace layout (fp32)
  float* ws   = (float*)d_ws;
  float* O    = ws;                                  // [N, VOCAB]  128 MB
  float* dinv = O    + (size_t)N * VOCABS;           // [N] (deg, then rsqrt(deg))
  float* h0   = dinv + N;                            // [N, HID]
  float* t    = h0   + (size_t)N * HID;              // [N, D1] pre-scatter features
  float* g    = t    + (size_t)N * D1;               // [N, D1] scatter accumulator
  float* h3   = g    + (size_t)N * D1;               // [N, HID]

  // ---- normalization: deg (incl. self loop) -> dinv = rsqrt(deg) ----
  fill_f32<<<cdiv_h(N, 256), 256, 0, stream>>>(dinv, 1.0f, N);
  count_deg<<<cdiv_h(E, 256), 256, 0, stream>>>(dstv, dinv, E);
  rsqrt_ip<<<cdiv_h(N, 256), 256, 0, stream>>>(dinv, N);

  // ---- embedding lookup ----
  embed_k<<<cdiv_h((long long)N * HID, 256), 256, 0, stream>>>(x, emb, h0, N);

  // ---- GCN conv 1: relu(scatter(norm * (h0@W1)[src]) + self + b1) ----
  gemm_bias<<<cdiv_h((long long)N * D1, 256), 256, 0, stream>>>(h0, W1, nullptr, t, N, HID, D1, 0);
  fill_f32<<<cdiv_h((long long)N * D1, 256), 256, 0, stream>>>(g, 0.0f, (long long)N * D1);
  scatter_edges<<<cdiv_h((long long)E * D1, 256), 256, 0, stream>>>(srcv, dstv, dinv, t, g, E, D1);
  finalize_conv<<<cdiv_h((long long)N * D1, 256), 256, 0, stream>>>(g, t, dinv, b1, N, D1, 1);

  // ---- GCN conv 2 ----
  gemm_bias<<<cdiv_h((long long)N * D1, 256), 256, 0, stream>>>(g, W2, nullptr, t, N, D1, D1, 0);
  fill_f32<<<cdiv_h((long long)N * D1, 256), 256, 0, stream>>>(g, 0.0f, (long long)N * D1);
  scatter_edges<<<cdiv_h((long long)E * D1, 256), 256, 0, stream>>>(srcv, dstv, dinv, t, g, E, D1);
  finalize_conv<<<cdiv_h((long long)N * D1, 256), 256, 0, stream>>>(g, t, dinv, b2, N, D1, 1);

  // ---- dense head: h3 = g@W3 + b3 ; O = h3@Wd + bd ----
  gemm_bias<<<cdiv_h((long long)N * HID, 256), 256, 0, stream>>>(g, W3, b3, h3, N, D1, HID, 0);
  gemm_bias<<<cdiv_h((long long)N * VOCABS, 256), 256, 0, stream>>>(h3, Wd, bd, O, N, HID, VOCABS, 0);

  // ---- S = sigmoid(O @ O^T) via fp32 WMMA ----
  dim3 grid(GN / BN, GN / BM);
  gram_sigmoid_wmma<<<grid, 256, 0, stream>>>(O, S);
}